// ContrastMemory_v2_15453292331568
// MI455X (gfx1250) — compile-verified
//
#include <hip/hip_runtime.h>

typedef float v2f __attribute__((ext_vector_type(2)));
typedef float v8f __attribute__((ext_vector_type(8)));
typedef float f4  __attribute__((ext_vector_type(4)));

#define B_    128
#define D_    128
#define OUTN  600000
#define K_    8192
#define P_    150
#define P2_   10
#define KP_   (K_ + P_)   /* 8342 */
#define NO_   (P2_ + K_)  /* 8202 */
#define TINV  14.285714285714286f
#define MOM_  0.5f

// Placeholder for host-side MID_IDX = rng(0).integers(50,100,10) (fixed constants
// in the reference; not reproducible without numpy PCG64). Values in [50,100).
// Slot 0 is overwritten with index 0 downstream, matching the reference.
__device__ __constant__ int c_mid[P2_] = {85, 63, 90, 77, 51, 96, 68, 59, 73, 82};

// ---------------------------------------------------------------------------
// Kernel A: NCE dot products via V_WMMA_F32_16X16X4_F32.
// One wave handles a 16-row k-tile of one batch element b.
// Chain over D=128 in K=4 steps; B-operand broadcasts v across all 16 columns.
// e1[b][k] = exp(dot(mem2[idx[b,k]], v1[b]) / T)   (feeds out_v1)
// e2[b][k] = exp(dot(mem1[idx[b,k]], v2[b]) / T)   (feeds out_v2)
// ---------------------------------------------------------------------------
__global__ void dots_wmma_kernel(const float* __restrict__ mem1,
                                 const float* __restrict__ mem2,
                                 const float* __restrict__ v1,
                                 const float* __restrict__ v2,
                                 const int*   __restrict__ idx,
                                 float* __restrict__ e1,
                                 float* __restrict__ e2)
{
    __shared__ float sv1[D_], sv2[D_];
    const int b   = blockIdx.x;
    const int tid = threadIdx.x;                 // 128 threads = 4 waves
    if (tid < D_) { sv1[tid] = v1[b * D_ + tid]; sv2[tid] = v2[b * D_ + tid]; }
    __syncthreads();

    const int wave = tid >> 5;
    const int lane = tid & 31;
    const int k0   = (blockIdx.y * 4 + wave) * 16;
    if (k0 >= KP_) return;                       // wave-uniform exit: EXEC stays full for WMMA

    const int  m    = lane & 15;                 // A-matrix row within tile (lanes 0-15 / 16-31)
    const int  krow = k0 + m;
    const int  row  = (krow < KP_) ? idx[b * KP_ + krow] : 0;
    const float* r1 = mem1 + (long)row * D_;
    const float* r2 = mem2 + (long)row * D_;
    const int  koff = (lane >= 16) ? 2 : 0;      // A layout: VGPR0 holds K=0 (lo lanes) / K=2 (hi)

    v8f c1 = {}; v8f c2 = {};
    #pragma unroll 8
    for (int kk = 0; kk < D_; kk += 4) {
        const int p = kk + koff;
        v2f a1; a1.x = r1[p];  a1.y = r1[p + 1];   // A tile from mem1 rows
        v2f a2; a2.x = r2[p];  a2.y = r2[p + 1];   // A tile from mem2 rows
        v2f b1; b1.x = sv1[p]; b1.y = sv1[p + 1];  // B = v1 broadcast across N
        v2f b2; b2.x = sv2[p]; b2.y = sv2[p + 1];  // B = v2 broadcast across N
        c2 = __builtin_amdgcn_wmma_f32_16x16x4_f32(false, a1, false, b2, (short)0, c2, false, false);
        c1 = __builtin_amdgcn_wmma_f32_16x16x4_f32(false, a2, false, b1, (short)0, c1, false, false);
    }

    // C layout: VGPR j = row (j | (lane>=16 ? 8 : 0)), N = lane & 15. N identical -> read N=0.
    if (m == 0) {
        const int mb = (lane >= 16) ? 8 : 0;
        #pragma unroll
        for (int j = 0; j < 8; ++j) {
            const int kr = k0 + mb + j;
            if (kr < KP_) {
                e1[b * KP_ + kr] = __expf(c1[j] * TINV);
                e2[b * KP_ + kr] = __expf(c2[j] * TINV);
            }
        }
    }
}

// ---------------------------------------------------------------------------
// Kernel B: teacher/student relation gap for k < P (needs row norms -> VALU).
// diff[b][k] = dot(w1^,v1^) - dot(w2^,v2^)
// ---------------------------------------------------------------------------
__global__ void rel_kernel(const float* __restrict__ mem1,
                           const float* __restrict__ mem2,
                           const float* __restrict__ v1,
                           const float* __restrict__ v2,
                           const int*   __restrict__ idx,
                           float* __restrict__ diff)
{
    __shared__ float sv1[D_], sv2[D_];
    __shared__ float sinv1, sinv2;
    const int b = blockIdx.x;
    const int t = threadIdx.x;                   // 192 threads
    if (t < D_) { sv1[t] = v1[b * D_ + t]; sv2[t] = v2[b * D_ + t]; }
    __syncthreads();
    if (t == 0) {
        float n1 = 0.f, n2 = 0.f;
        for (int d = 0; d < D_; ++d) { n1 += sv1[d] * sv1[d]; n2 += sv2[d] * sv2[d]; }
        sinv1 = rsqrtf(n1); sinv2 = rsqrtf(n2);
    }
    __syncthreads();
    if (t < P_) {
        const int row   = idx[b * KP_ + t];
        const float* r1 = mem1 + (long)row * D_;
        const float* r2 = mem2 + (long)row * D_;
        float d1 = 0.f, n1 = 0.f, d2 = 0.f, n2 = 0.f;
        for (int d = 0; d < D_; ++d) {
            const float w1 = r1[d], w2 = r2[d];
            d1 += w1 * sv1[d]; n1 += w1 * w1;
            d2 += w2 * sv2[d]; n2 += w2 * w2;
        }
        diff[b * P_ + t] = d1 * rsqrtf(n1) * sinv1 - d2 * rsqrtf(n2) * sinv2;
    }
}

// ---------------------------------------------------------------------------
// Kernel C: stable descending rank over P=150, pick ranks MID_IDX, force sel[0]=0.
// Replicates jnp.argsort(-diff) (stable: ties broken by lower index).
// ---------------------------------------------------------------------------
__global__ void select_kernel(const float* __restrict__ diff, int* __restrict__ sel)
{
    __shared__ float sd[P_];
    const int b = blockIdx.x;
    const int t = threadIdx.x;                   // 160 threads
    if (t < P_) sd[t] = diff[b * P_ + t];
    __syncthreads();
    if (t < P_) {
        const float v = sd[t];
        int rank = 0;
        for (int i = 0; i < P_; ++i) {
            const float u = sd[i];
            rank += (u > v) || (u == v && i < t);
        }
        #pragma unroll
        for (int j = 0; j < P2_; ++j)
            if (rank == c_mid[j]) sel[b * P2_ + j] = t;
    }
    __syncthreads();
    if (t == 0) sel[b * P2_ + 0] = 0;
}

// ---------------------------------------------------------------------------
// Kernel D: assemble [gathered P2 | columns P..P+K] into d_out, per-b partial sums.
// ---------------------------------------------------------------------------
__global__ void assemble_kernel(const float* __restrict__ e1,
                                const float* __restrict__ e2,
                                const int*   __restrict__ sel,
                                float* __restrict__ out1,
                                float* __restrict__ out2,
                                float* __restrict__ part)
{
    const int b = blockIdx.x;
    const int t = threadIdx.x;                   // 256 threads
    float s1 = 0.f, s2 = 0.f;
    for (int i = t; i < NO_; i += 256) {
        const int src = (i < P2_) ? sel[b * P2_ + i] : (P_ + (i - P2_));
        const float a = e1[b * KP_ + src];
        const float c = e2[b * KP_ + src];
        out1[b * NO_ + i] = a;
        out2[b * NO_ + i] = c;
        s1 += a; s2 += c;
    }
    __shared__ float r1[256], r2[256];
    r1[t] = s1; r2[t] = s2; __syncthreads();
    for (int o = 128; o > 0; o >>= 1) {
        if (t < o) { r1[t] += r1[t + o]; r2[t] += r2[t + o]; }
        __syncthreads();
    }
    if (t == 0) { part[b] = r1[0]; part[B_ + b] = r2[0]; }
}

// ---------------------------------------------------------------------------
// Kernel E: reduce partials -> 1/Z1, 1/Z2 (fixed-order, deterministic).
// ---------------------------------------------------------------------------
__global__ void zfinal_kernel(const float* __restrict__ part, float* __restrict__ inv)
{
    const int t = threadIdx.x;                   // 128 threads
    __shared__ float r1[B_], r2[B_];
    r1[t] = part[t]; r2[t] = part[B_ + t]; __syncthreads();
    for (int o = 64; o > 0; o >>= 1) {
        if (t < o) { r1[t] += r1[t + o]; r2[t] += r2[t + o]; }
        __syncthreads();
    }
    if (t == 0) {
        const float cnt = (float)(B_ * NO_);
        inv[0] = cnt / (r1[0] * (float)OUTN);    // 1/Z1 = count / (sum * OUT)
        inv[1] = cnt / (r2[0] * (float)OUTN);
    }
}

// ---------------------------------------------------------------------------
// Kernel F: scale both output score arrays in place.
// ---------------------------------------------------------------------------
__global__ void scale_kernel(float* __restrict__ out1, float* __restrict__ out2,
                             const float* __restrict__ inv)
{
    const float i1 = inv[0], i2 = inv[1];
    const long  n  = (long)B_ * NO_;
    const long  stride = (long)gridDim.x * blockDim.x;
    for (long i = (long)blockIdx.x * blockDim.x + threadIdx.x; i < n; i += stride) {
        out1[i] *= i1;
        out2[i] *= i2;
    }
}

// ---------------------------------------------------------------------------
// Kernel G: 614 MB streaming passthrough of the memory banks.
// Non-temporal f4 copies (no L2 residency) + prefetch hint (global_prefetch_b8).
// ---------------------------------------------------------------------------
__global__ void copy_nt_kernel(const f4* __restrict__ src, f4* __restrict__ dst, long n4)
{
    const long stride = (long)gridDim.x * blockDim.x;
    for (long i = (long)blockIdx.x * blockDim.x + threadIdx.x; i < n4; i += stride) {
        __builtin_prefetch(src + i + stride, 0, 1);
        __builtin_nontemporal_store(__builtin_nontemporal_load(src + i), dst + i);
    }
}

// ---------------------------------------------------------------------------
// Kernel H: EMA + L2-normalize the B updated rows (after the copy).
// Deterministic last-write-wins on duplicate y (matches .at[y].set semantics).
// ---------------------------------------------------------------------------
__global__ void update_rows_kernel(const float* __restrict__ mem1,
                                   const float* __restrict__ mem2,
                                   const float* __restrict__ v1,
                                   const float* __restrict__ v2,
                                   const int*   __restrict__ y,
                                   float* __restrict__ nm1,
                                   float* __restrict__ nm2)
{
    const int b = blockIdx.x;
    const int d = threadIdx.x;                   // 128 threads = D = 4 waves
    const int row = y[b];
    for (int bb = b + 1; bb < B_; ++bb)          // uniform across block
        if (y[bb] == row) return;

    const float p1 = mem1[(long)row * D_ + d] * MOM_ + v1[b * D_ + d] * (1.0f - MOM_);
    const float p2 = mem2[(long)row * D_ + d] * MOM_ + v2[b * D_ + d] * (1.0f - MOM_);

    float q1 = p1 * p1, q2 = p2 * p2;
    for (int o = 16; o > 0; o >>= 1) {           // wave32 butterfly
        q1 += __shfl_xor(q1, o);
        q2 += __shfl_xor(q2, o);
    }
    __shared__ float s1[4], s2[4];
    const int wv = d >> 5, ln = d & 31;
    if (ln == 0) { s1[wv] = q1; s2[wv] = q2; }
    __syncthreads();
    const float n1 = s1[0] + s1[1] + s1[2] + s1[3];
    const float n2 = s2[0] + s2[1] + s2[2] + s2[3];
    nm1[(long)row * D_ + d] = p1 * rsqrtf(n1);
    nm2[(long)row * D_ + d] = p2 * rsqrtf(n2);
}

// ---------------------------------------------------------------------------
extern "C" void kernel_launch(void* const* d_in, const int* in_sizes, int n_in,
                              void* d_out, int out_size, void* d_ws, size_t ws_size,
                              hipStream_t stream)
{
    // Inputs (setup_inputs order): epoch, v1, v2, y, idx, memory_v1, memory_v2
    const float* v1   = (const float*)d_in[1];
    const float* v2   = (const float*)d_in[2];
    const int*   y    = (const int*)  d_in[3];
    const int*   idx  = (const int*)  d_in[4];
    const float* mem1 = (const float*)d_in[5];
    const float* mem2 = (const float*)d_in[6];

    // Outputs: out_v1 [B,NO], out_v2 [B,NO], new_mem_v1 [OUT,D], new_mem_v2 [OUT,D]
    float* out  = (float*)d_out;
    float* out1 = out;
    float* out2 = out1 + (long)B_ * NO_;
    float* nm1  = out2 + (long)B_ * NO_;
    float* nm2  = nm1  + (long)OUTN * D_;

    // Workspace layout
    float* w    = (float*)d_ws;
    float* e1   = w;                             // B*KP
    float* e2   = e1 + (long)B_ * KP_;           // B*KP
    float* diff = e2 + (long)B_ * KP_;           // B*P
    float* part = diff + (long)B_ * P_;          // 2*B
    float* inv  = part + 2 * B_;                 // 2 (+pad)
    int*   sel  = (int*)(inv + 8);               // B*P2

    // A: WMMA dot products over all K+P columns
    {
        const int ktiles = (KP_ + 15) / 16;      // 522
        const int gy     = (ktiles + 3) / 4;     // 131 (4 waves/block)
        dots_wmma_kernel<<<dim3(B_, gy), 128, 0, stream>>>(mem1, mem2, v1, v2, idx, e1, e2);
    }
    // B: relation gap for k < P
    rel_kernel<<<B_, 192, 0, stream>>>(mem1, mem2, v1, v2, idx, diff);
    // C: rank-select MID_IDX positions
    select_kernel<<<B_, 160, 0, stream>>>(diff, sel);
    // D: assemble outputs + per-b partial sums
    assemble_kernel<<<B_, 256, 0, stream>>>(e1, e2, sel, out1, out2, part);
    // E: Z normalizers
    zfinal_kernel<<<1, B_, 0, stream>>>(part, inv);
    // F: scale outputs by 1/Z
    scale_kernel<<<1024, 256, 0, stream>>>(out1, out2, inv);
    // G: stream both memory banks to the output (non-temporal)
    {
        const long n4 = (long)OUTN * D_ / 4;     // 19.2M f4 per bank
        copy_nt_kernel<<<4096, 256, 0, stream>>>((const f4*)mem1, (f4*)nm1, n4);
        copy_nt_kernel<<<4096, 256, 0, stream>>>((const f4*)mem2, (f4*)nm2, n4);
    }
    // H: EMA + renormalize the B touched rows (after copy; same stream orders it)
    update_rows_kernel<<<B_, D_, 0, stream>>>(mem1, mem2, v1, v2, y, nm1, nm2);
}